// MDDGCN_45878840656368
// MI455X (gfx1250) — compile-verified
//
#include <hip/hip_runtime.h>
#include <math.h>

typedef __attribute__((ext_vector_type(2))) float v2f;
typedef __attribute__((ext_vector_type(8))) float v8f;

// ---------------------------------------------------------------------------
// WMMA fp32 GEMM: Y[M,C] = X[M,K] @ W[K,C] (+bias) (+relu)
// One wave32 computes a 16x64 strip: one A fragment feeds 4 accumulators
// (4 x V_WMMA_F32_16X16X4_F32 per k-step). block = 256 threads = 8 waves
// covering 128 rows x 64 cols. grid = (ceil(M/128), C/64).
// ---------------------------------------------------------------------------
__global__ __launch_bounds__(256) void gemm_wmma_f32(
    const float* __restrict__ X, const float* __restrict__ W,
    const float* __restrict__ bias, float* __restrict__ Y,
    int M, int K, int C, int do_relu)
{
    const int wave = threadIdx.x >> 5;                 // 0..7: row tile in block
    const int lane = threadIdx.x & 31;
    const int hi   = lane >> 4;                        // lane half: K pair 2,3
    const int lm   = lane & 15;
    const int row0 = (blockIdx.x * 8 + wave) << 4;
    const int col0 = blockIdx.y << 6;                  // 64 cols per wave

    int rA = row0 + lm; if (rA > M - 1) rA = M - 1;    // clamp (keeps EXEC full)
    const float* __restrict__ xrow = X + (size_t)rA * K;

    v8f acc0 = {0.f,0.f,0.f,0.f,0.f,0.f,0.f,0.f};
    v8f acc1 = acc0, acc2 = acc0, acc3 = acc0;

#pragma unroll 4
    for (int kk = 0; kk < K; kk += 4) {
        const int ka = kk + (hi << 1);
        v2f a; a.x = xrow[ka]; a.y = xrow[ka + 1];
        const float* __restrict__ w0p = W + (size_t)ka * C + col0 + lm;
        const float* __restrict__ w1p = w0p + C;       // row ka+1
        v2f b0 = {w0p[ 0], w1p[ 0]};
        v2f b1 = {w0p[16], w1p[16]};
        v2f b2 = {w0p[32], w1p[32]};
        v2f b3 = {w0p[48], w1p[48]};
        acc0 = __builtin_amdgcn_wmma_f32_16x16x4_f32(false, a, false, b0, (short)0, acc0, false, false);
        acc1 = __builtin_amdgcn_wmma_f32_16x16x4_f32(false, a, false, b1, (short)0, acc1, false, false);
        acc2 = __builtin_amdgcn_wmma_f32_16x16x4_f32(false, a, false, b2, (short)0, acc2, false, false);
        acc3 = __builtin_amdgcn_wmma_f32_16x16x4_f32(false, a, false, b3, (short)0, acc3, false, false);
    }

    v8f* accs[4] = {&acc0, &acc1, &acc2, &acc3};
#pragma unroll
    for (int t = 0; t < 4; ++t) {
        const int col = col0 + (t << 4) + lm;
        const float bv = bias ? bias[col] : 0.0f;
        const v8f av = *accs[t];
#pragma unroll
        for (int v = 0; v < 8; ++v) {
            const int row = row0 + v + (hi << 3);
            if (row < M) {
                float val = av[v] + bv;
                if (do_relu) val = fmaxf(val, 0.0f);
                Y[(size_t)row * C + col] = val;
            }
        }
    }
}

// ---------------------------------------------------------------------------
// Degree (with self loops) and symmetric-norm helpers
// ---------------------------------------------------------------------------
__global__ void deg_init(float* deg, int Nn) {
    int i = blockIdx.x * blockDim.x + threadIdx.x;
    if (i < Nn) deg[i] = 1.0f;                       // self-loop contribution
}
__global__ void deg_count(const int* __restrict__ dst, int E, float* deg) {
    int e = blockIdx.x * blockDim.x + threadIdx.x;
    if (e < E) atomicAdd(&deg[dst[e]], 1.0f);
}
__global__ void deg_rsqrt(float* deg, int Nn) {
    int i = blockIdx.x * blockDim.x + threadIdx.x;
    if (i < Nn) deg[i] = rsqrtf(deg[i]);             // deg >= 1 always
}

// ---------------------------------------------------------------------------
// Attention: wts = softmax(h @ att_w2 + b2); xa = x * repeat(wts,(64,64))
// One wave32 per node; shuffle reduction (warpSize==32 on gfx1250).
// ---------------------------------------------------------------------------
__global__ void att_kernel(const float* __restrict__ x, const float* __restrict__ h,
                           const float* __restrict__ w2, const float* __restrict__ b2,
                           float* __restrict__ xa, int Nn)
{
    const int node = blockIdx.x * (blockDim.x >> 5) + (threadIdx.x >> 5);
    const int lane = threadIdx.x & 31;
    if (node >= Nn) return;
    const float* hrow = h + (size_t)node * 128;
    float d0 = 0.f, d1 = 0.f;
#pragma unroll
    for (int f = lane; f < 128; f += 32) {
        const float hv = hrow[f];
        d0 += hv * w2[f * 2 + 0];
        d1 += hv * w2[f * 2 + 1];
    }
    for (int off = 16; off; off >>= 1) {
        d0 += __shfl_down(d0, off, 32);
        d1 += __shfl_down(d1, off, 32);
    }
    d0 = __shfl(d0, 0, 32) + b2[0];
    d1 = __shfl(d1, 0, 32) + b2[1];
    const float m  = fmaxf(d0, d1);
    const float e0 = __expf(d0 - m), e1 = __expf(d1 - m);
    const float inv = 1.0f / (e0 + e1);
    const float w0 = e0 * inv, w1 = e1 * inv;
    const float* xrow = x  + (size_t)node * 128;
    float*       orow = xa + (size_t)node * 128;
#pragma unroll
    for (int f = lane; f < 128; f += 32)
        orow[f] = xrow[f] * (f < 64 ? w0 : w1);
}

// ---------------------------------------------------------------------------
// GCN aggregation: out = selfloop + sum_e norm(e) * h[src(e)]
// selfloop_init fully overwrites out (no memset needed over poisoned ws).
// ---------------------------------------------------------------------------
__global__ void selfloop_init(const float* __restrict__ h, const float* __restrict__ dinv,
                              float* __restrict__ out, int Nn, int C)
{
    const size_t i = (size_t)blockIdx.x * blockDim.x + threadIdx.x;
    if (i >= (size_t)Nn * C) return;
    const int n = (int)(i / C);
    const float dv = dinv[n];
    out[i] = h[i] * dv * dv;
}

__global__ void scatter_kernel(const float* __restrict__ h,
                               const int* __restrict__ src, const int* __restrict__ dst,
                               const float* __restrict__ dinv, float* __restrict__ out,
                               int E, int C)
{
    const int e    = blockIdx.x * (blockDim.x >> 5) + (threadIdx.x >> 5);
    const int lane = threadIdx.x & 31;
    if (e >= E) return;
    const int s = src[e], d = dst[e];
    const float norm = dinv[s] * dinv[d];
    const float* hrow = h   + (size_t)s * C;
    float*       orow = out + (size_t)d * C;
    for (int f = lane; f < C; f += 32)
        atomicAdd(&orow[f], hrow[f] * norm);
}

// ---------------------------------------------------------------------------
// BatchNorm (training-mode batch statistics, biased var)
// ---------------------------------------------------------------------------
__global__ __launch_bounds__(256) void bn_stats(const float* __restrict__ y, int Nn, int C,
                                                float* __restrict__ mean, float* __restrict__ rstd)
{
    const int c = blockIdx.x;
    float s = 0.f, q = 0.f;
    for (int i = threadIdx.x; i < Nn; i += blockDim.x) {
        const float v = y[(size_t)i * C + c];
        s += v; q += v * v;
    }
    __shared__ float sh_s[256], sh_q[256];
    sh_s[threadIdx.x] = s; sh_q[threadIdx.x] = q;
    __syncthreads();
    for (int off = 128; off; off >>= 1) {
        if ((int)threadIdx.x < off) {
            sh_s[threadIdx.x] += sh_s[threadIdx.x + off];
            sh_q[threadIdx.x] += sh_q[threadIdx.x + off];
        }
        __syncthreads();
    }
    if (threadIdx.x == 0) {
        const float m = sh_s[0] / (float)Nn;
        const float var = sh_q[0] / (float)Nn - m * m;
        mean[c] = m;
        rstd[c] = rsqrtf(var + 1e-5f);
    }
}

__global__ void bn_apply(const float* __restrict__ y, const float* __restrict__ mean,
                         const float* __restrict__ rstd, const float* __restrict__ g,
                         const float* __restrict__ b, const float* __restrict__ resid,
                         float* __restrict__ out, int Nn, int C, int do_relu)
{
    const size_t i = (size_t)blockIdx.x * blockDim.x + threadIdx.x;
    if (i >= (size_t)Nn * C) return;
    const int c = (int)(i & (size_t)(C - 1));         // C is a power of two
    float v = g[c] * (y[i] - mean[c]) * rstd[c] + b[c];
    if (do_relu) v = fmaxf(v, 0.f);
    if (resid)   v += resid[i];
    out[i] = v;
}

// ---------------------------------------------------------------------------
// Gated fusion: gate = sigmoid([a, r] @ gw + gb); a = gate*a + (1-gate)*r
// One wave32 per node; in-place on a.
// ---------------------------------------------------------------------------
__global__ void gate_kernel(float* __restrict__ a, const float* __restrict__ r,
                            const float* __restrict__ gw, const float* __restrict__ gb,
                            int Nn, int C)
{
    const int node = blockIdx.x * (blockDim.x >> 5) + (threadIdx.x >> 5);
    const int lane = threadIdx.x & 31;
    if (node >= Nn) return;
    float*       arow = a + (size_t)node * C;
    const float* rrow = r + (size_t)node * C;
    float dot = 0.f;
    for (int f = lane; f < C; f += 32)
        dot += arow[f] * gw[f] + rrow[f] * gw[C + f];
    for (int off = 16; off; off >>= 1) dot += __shfl_down(dot, off, 32);
    dot = __shfl(dot, 0, 32) + gb[0];
    const float gate = 1.0f / (1.0f + __expf(-dot));
    for (int f = lane; f < C; f += 32)
        arow[f] = gate * arow[f] + (1.0f - gate) * rrow[f];
}

// ---------------------------------------------------------------------------
// Classifier tail: out[n] = t[n,:] @ w[64] + b  (wave32 per node)
// ---------------------------------------------------------------------------
__global__ void final_dot(const float* __restrict__ t, const float* __restrict__ w,
                          const float* __restrict__ b, float* __restrict__ out, int Nn)
{
    const int node = blockIdx.x * (blockDim.x >> 5) + (threadIdx.x >> 5);
    const int lane = threadIdx.x & 31;
    if (node >= Nn) return;
    const float* row = t + (size_t)node * 64;
    float d = row[lane] * w[lane] + row[lane + 32] * w[lane + 32];
    for (int off = 16; off; off >>= 1) d += __shfl_down(d, off, 32);
    if (lane == 0) out[node] = d + b[0];
}

// ---------------------------------------------------------------------------
extern "C" void kernel_launch(void* const* d_in, const int* in_sizes, int n_in,
                              void* d_out, int out_size, void* d_ws, size_t ws_size,
                              hipStream_t stream)
{
    (void)n_in; (void)out_size; (void)ws_size;

    const float* x      = (const float*)d_in[0];
    const int*   ei     = (const int*)  d_in[1];
    const float* att_w1 = (const float*)d_in[2];
    const float* att_b1 = (const float*)d_in[3];
    const float* att_w2 = (const float*)d_in[4];
    const float* att_b2 = (const float*)d_in[5];
    const float* c1_w   = (const float*)d_in[6];
    const float* c1_b   = (const float*)d_in[7];
    const float* bn1_g  = (const float*)d_in[8];
    const float* bn1_b  = (const float*)d_in[9];
    const float* c2_w   = (const float*)d_in[10];
    const float* c2_b   = (const float*)d_in[11];
    const float* bn2_g  = (const float*)d_in[12];
    const float* bn2_b  = (const float*)d_in[13];
    const float* c3_w   = (const float*)d_in[14];
    const float* c3_b   = (const float*)d_in[15];
    const float* bn3_g  = (const float*)d_in[16];
    const float* bn3_b  = (const float*)d_in[17];
    const float* fcm_w  = (const float*)d_in[18];
    const float* fcm_b  = (const float*)d_in[19];
    const float* fca_w  = (const float*)d_in[20];
    const float* fca_b  = (const float*)d_in[21];
    const float* gm_w   = (const float*)d_in[22];
    const float* gm_b   = (const float*)d_in[23];
    const float* ga_w   = (const float*)d_in[24];
    const float* ga_b   = (const float*)d_in[25];
    const float* cl_w1  = (const float*)d_in[26];
    const float* cl_b1  = (const float*)d_in[27];
    const float* cl_w2  = (const float*)d_in[28];
    const float* cl_b2  = (const float*)d_in[29];

    const int Nn = in_sizes[0] / 128;     // 50000
    const int E  = in_sizes[1] / 2;       // 800000
    const int* srcE = ei;
    const int* dstE = ei + E;

    // workspace layout (~128.2 MB of fp32)
    float* B0   = (float*)d_ws;                    // N x 128 scratch (gemm out)
    float* B1   = B0 + (size_t)Nn * 128;           // xa, later x3
    float* B2   = B1 + (size_t)Nn * 128;           // aggregation target
    float* B3   = B2 + (size_t)Nn * 128;           // x_after1 / xo
    float* B4   = B3 + (size_t)Nn * 128;           // main_res / aux_res
    float* deg  = B4 + (size_t)Nn * 128;           // N   (becomes d^{-1/2})
    float* mean = deg + Nn;                        // 128
    float* rstd = mean + 128;                      // 128

    const int TPB = 256;
    dim3 blk(TPB);
    auto cdiv = [](int a, int b) { return (a + b - 1) / b; };
    const dim3 gemmB(256);                         // 8 waves x 16 rows
    const int rowT = cdiv(Nn, 128);                // 128 rows per block

    // --- degree / symmetric norm ---------------------------------------
    deg_init <<<cdiv(Nn, TPB), blk, 0, stream>>>(deg, Nn);
    deg_count<<<cdiv(E,  TPB), blk, 0, stream>>>(dstE, E, deg);
    deg_rsqrt<<<cdiv(Nn, TPB), blk, 0, stream>>>(deg, Nn);

    // --- attention ------------------------------------------------------
    gemm_wmma_f32<<<dim3(rowT, 2), gemmB, 0, stream>>>(x, att_w1, att_b1, B0, Nn, 128, 128, 1);
    att_kernel   <<<cdiv(Nn, 8), blk, 0, stream>>>(x, B0, att_w2, att_b2, B1, Nn);

    // --- block 1: x1 = relu(bn(gcn(xa))) + xa --------------------------
    gemm_wmma_f32<<<dim3(rowT, 2), gemmB, 0, stream>>>(B1, c1_w, c1_b, B0, Nn, 128, 128, 0);
    selfloop_init<<<cdiv(Nn * 128, TPB), blk, 0, stream>>>(B0, deg, B2, Nn, 128);
    scatter_kernel<<<cdiv(E, 8), blk, 0, stream>>>(B0, srcE, dstE, deg, B2, E, 128);
    bn_stats<<<128, blk, 0, stream>>>(B2, Nn, 128, mean, rstd);
    bn_apply<<<cdiv(Nn * 128, TPB), blk, 0, stream>>>(B2, mean, rstd, bn1_g, bn1_b, B1, B3, Nn, 128, 1);

    // --- block 2: xo = relu(bn(gcn(x1))) + x1 --------------------------
    gemm_wmma_f32<<<dim3(rowT, 2), gemmB, 0, stream>>>(B3, c2_w, c2_b, B0, Nn, 128, 128, 0);
    selfloop_init<<<cdiv(Nn * 128, TPB), blk, 0, stream>>>(B0, deg, B2, Nn, 128);
    scatter_kernel<<<cdiv(E, 8), blk, 0, stream>>>(B0, srcE, dstE, deg, B2, E, 128);
    bn_stats<<<128, blk, 0, stream>>>(B2, Nn, 128, mean, rstd);
    bn_apply<<<cdiv(Nn * 128, TPB), blk, 0, stream>>>(B2, mean, rstd, bn2_g, bn2_b, B3, B3, Nn, 128, 1);

    // --- main branch + gated fusion ------------------------------------
    gemm_wmma_f32<<<dim3(rowT, 2), gemmB, 0, stream>>>(x, fcm_w, fcm_b, B4, Nn, 128, 128, 1);
    gate_kernel<<<cdiv(Nn, 8), blk, 0, stream>>>(B3, B4, gm_w, gm_b, Nn, 128);

    // --- block 3: x3 = bn(gcn(xo)) (no relu) ---------------------------
    gemm_wmma_f32<<<dim3(rowT, 1), gemmB, 0, stream>>>(B3, c3_w, c3_b, B0, Nn, 128, 64, 0);
    selfloop_init<<<cdiv(Nn * 64, TPB), blk, 0, stream>>>(B0, deg, B2, Nn, 64);
    scatter_kernel<<<cdiv(E, 8), blk, 0, stream>>>(B0, srcE, dstE, deg, B2, E, 64);
    bn_stats<<<64, blk, 0, stream>>>(B2, Nn, 64, mean, rstd);
    bn_apply<<<cdiv(Nn * 64, TPB), blk, 0, stream>>>(B2, mean, rstd, bn3_g, bn3_b, nullptr, B1, Nn, 64, 0);

    // --- aux branch + gated fusion -------------------------------------
    gemm_wmma_f32<<<dim3(rowT, 1), gemmB, 0, stream>>>(x, fca_w, fca_b, B4, Nn, 128, 64, 1);
    gate_kernel<<<cdiv(Nn, 8), blk, 0, stream>>>(B1, B4, ga_w, ga_b, Nn, 64);

    // --- classifier -----------------------------------------------------
    gemm_wmma_f32<<<dim3(rowT, 1), gemmB, 0, stream>>>(B1, cl_w1, cl_b1, B0, Nn, 64, 64, 1);
    final_dot<<<cdiv(Nn, 8), blk, 0, stream>>>(B0, cl_w2, cl_b2, (float*)d_out, Nn);
}